// EarthMoverDistance_5007931867610
// MI455X (gfx1250) — compile-verified
//
#include <hip/hip_runtime.h>

typedef float v2f __attribute__((ext_vector_type(2)));
typedef float v8f __attribute__((ext_vector_type(8)));

#define NSTATES 64
#define SINK_ITERS 50
#define EPSI 0.1f
#define TINYF 1e-8f

// 16-lane butterfly sum (stays within each half of a wave32)
__device__ __forceinline__ float hsum16(float x) {
#pragma unroll
    for (int m = 1; m < 16; m <<= 1) x += __shfl_xor(x, m, 32);
    return x;
}

// Block = 128 threads = 4 waves. Each wave processes 16 batch rows.
// LDS: KB = K in WMMA-B fragment layout, MB = (K*C) fragments, stage = per-wave
// 16x64 f32 transpose staging.
__global__ __launch_bounds__(128)
void sinkhorn_emd_kernel(const float* __restrict__ p,
                         const float* __restrict__ q,
                         const float* __restrict__ cost,
                         float* __restrict__ out) {
    __shared__ v2f   KB[16 * 4 * 32];     // 16 KB: [kchunk][ntile][lane]
    __shared__ v2f   MB[16 * 4 * 32];     // 16 KB: K .* C fragments
    __shared__ float stage[4][16 * 64];   // 16 KB: per-wave D->A relayout

    const int tid  = threadIdx.x;
    const int wave = tid >> 5;
    const int lane = tid & 31;
    const int half = lane >> 4;   // 0: lanes 0-15, 1: lanes 16-31
    const int l15  = lane & 15;

    // ---- Bake K and K*C into WMMA B-matrix fragment layout in LDS ----
    // B tile for (kchunk c, ntile t): VGPR0 = K rows {4c (lanes 0-15), 4c+2 (16-31)},
    // VGPR1 = rows {4c+1, 4c+3}; column = 16t + (lane&15).
    for (int e = tid; e < 16 * 4 * 32; e += 128) {
        int c   = e >> 7;
        int t   = (e >> 5) & 3;
        int ln  = e & 31;
        int kr  = 4 * c + ((ln >> 4) << 1);
        int col = 16 * t + (ln & 15);
        float c0 = cost[kr * NSTATES + col];
        float c1 = cost[(kr + 1) * NSTATES + col];
        float k0 = __expf(-c0 * (1.0f / EPSI));
        float k1 = __expf(-c1 * (1.0f / EPSI));
        KB[e] = v2f{k0, k1};
        MB[e] = v2f{k0 * c0, k1 * c1};
    }
    __syncthreads();

    const int rowbase = blockIdx.x * 64 + wave * 16;

    // ---- Load p,q tiles directly in WMMA C/D layout, normalize rows ----
    // D layout: vgpr j of lane -> row (j + half*8), col (t*16 + l15)
    float qD[4][8], pD[4][8];
#pragma unroll
    for (int t = 0; t < 4; ++t) {
#pragma unroll
        for (int j = 0; j < 8; ++j) {
            int r = rowbase + j + half * 8;
            int cidx = t * 16 + l15;
            qD[t][j] = q[r * NSTATES + cidx] + TINYF;
            pD[t][j] = p[r * NSTATES + cidx] + TINYF;
        }
    }
#pragma unroll
    for (int j = 0; j < 8; ++j) {
        float sq = qD[0][j] + qD[1][j] + qD[2][j] + qD[3][j];
        float sp = pD[0][j] + pD[1][j] + pD[2][j] + pD[3][j];
        sq = hsum16(sq);
        sp = hsum16(sp);
        float rq = __builtin_amdgcn_rcpf(sq);
        float rp = __builtin_amdgcn_rcpf(sp);
#pragma unroll
        for (int t = 0; t < 4; ++t) { qD[t][j] *= rq; pD[t][j] *= rp; }
    }

    // ---- Sinkhorn iterations ----
    // a[] holds the current iterate in WMMA A layout:
    // chunk c: lane<16 -> {x[l15][4c], x[l15][4c+1]}, lane>=16 -> {x[l15][4c+2], x[l15][4c+3]}
    v2f a[16];
#pragma unroll
    for (int c = 0; c < 16; ++c) a[c] = v2f{1.0f, 1.0f};

    float* st = stage[wave];
    float vD[4][8];

    for (int it = 0; it < SINK_ITERS; ++it) {
        // v = q / (u @ K)   (K symmetric)
#pragma unroll
        for (int t = 0; t < 4; ++t) {
            v8f acc = {0.f, 0.f, 0.f, 0.f, 0.f, 0.f, 0.f, 0.f};
#pragma unroll
            for (int c = 0; c < 16; ++c) {
                v2f b = KB[(c * 4 + t) * 32 + lane];
                acc = __builtin_amdgcn_wmma_f32_16x16x4_f32(
                    false, a[c], false, b, (short)0, acc, false, false);
            }
#pragma unroll
            for (int j = 0; j < 8; ++j) {
                float v = qD[t][j] * __builtin_amdgcn_rcpf(acc[j]);
                vD[t][j] = v;
                st[(j + half * 8) * 64 + t * 16 + l15] = v;
            }
        }
        // D layout -> A layout via per-wave LDS (same-wave DS ops are in-order)
#pragma unroll
        for (int c = 0; c < 16; ++c)
            a[c] = *(const v2f*)&st[l15 * 64 + 4 * c + half * 2];

        // u = p / (v @ K^T) = p / (v @ K)
#pragma unroll
        for (int t = 0; t < 4; ++t) {
            v8f acc = {0.f, 0.f, 0.f, 0.f, 0.f, 0.f, 0.f, 0.f};
#pragma unroll
            for (int c = 0; c < 16; ++c) {
                v2f b = KB[(c * 4 + t) * 32 + lane];
                acc = __builtin_amdgcn_wmma_f32_16x16x4_f32(
                    false, a[c], false, b, (short)0, acc, false, false);
            }
#pragma unroll
            for (int j = 0; j < 8; ++j) {
                float u = pD[t][j] * __builtin_amdgcn_rcpf(acc[j]);
                st[(j + half * 8) * 64 + t * 16 + l15] = u;
            }
        }
#pragma unroll
        for (int c = 0; c < 16; ++c)
            a[c] = *(const v2f*)&st[l15 * 64 + 4 * c + half * 2];
    }

    // ---- emd_b = sum_j (u @ (K.*C))_bj * v_bj ----
    float e[8] = {0.f, 0.f, 0.f, 0.f, 0.f, 0.f, 0.f, 0.f};
#pragma unroll
    for (int t = 0; t < 4; ++t) {
        v8f acc = {0.f, 0.f, 0.f, 0.f, 0.f, 0.f, 0.f, 0.f};
#pragma unroll
        for (int c = 0; c < 16; ++c) {
            v2f b = MB[(c * 4 + t) * 32 + lane];
            acc = __builtin_amdgcn_wmma_f32_16x16x4_f32(
                false, a[c], false, b, (short)0, acc, false, false);
        }
#pragma unroll
        for (int j = 0; j < 8; ++j) e[j] += acc[j] * vD[t][j];
    }
#pragma unroll
    for (int j = 0; j < 8; ++j) {
        float s = hsum16(e[j]);
        if (l15 == 0) out[rowbase + j + half * 8] = s;
    }
}

extern "C" void kernel_launch(void* const* d_in, const int* in_sizes, int n_in,
                              void* d_out, int out_size, void* d_ws, size_t ws_size,
                              hipStream_t stream) {
    const float* p    = (const float*)d_in[0];
    const float* q    = (const float*)d_in[1];
    const float* cost = (const float*)d_in[2];
    float* out = (float*)d_out;

    int rows = in_sizes[0] / NSTATES;     // 262144
    int blocks = rows / 64;               // 64 rows per block (4 waves x 16)
    sinkhorn_emd_kernel<<<blocks, 128, 0, stream>>>(p, q, cost, out);
}